// MultinomialKLDivergenceLoss_84482006712456
// MI455X (gfx1250) — compile-verified
//
#include <hip/hip_runtime.h>
#include <stdint.h>

// Problem geometry from the reference: [64, 64, 25000] f32.
#define V_LEN     25000
#define TILE      1024        // floats per tile per array (4 KB = 256 lanes x 16B)
#define TILE_B    4096u       // tile bytes
#define NT        24          // full tiles: 24*1024 = 24576
#define REM_OFF   24576       // ragged tail start (424 elements)
#define TPB       256         // 8 wave32 per block
#define PIPE      4           // 4 LDS buffers, issue 3 tiles ahead

// One b128 async copy: global (saddr + 32-bit lane offset) -> LDS (lane's slot).
// CDNA5 path: tracked with ASYNCcnt, no VGPR data in flight.
__device__ __forceinline__ void async_ld16(uint64_t base, unsigned voff, unsigned lds_addr)
{
    asm volatile("global_load_async_to_lds_b128 %0, %1, %2"
                 :
                 : "v"(lds_addr), "v"(voff), "s"(base)
                 : "memory");
}

__global__ __launch_bounds__(TPB) void kl_row_kernel(const float* __restrict__ p,
                                                     const float* __restrict__ q,
                                                     float* __restrict__ ws)
{
    __shared__ alignas(16) float ldsP[PIPE][TILE];
    __shared__ alignas(16) float ldsQ[PIPE][TILE];
    __shared__ float redS[TPB];
    __shared__ float redM[TPB];

    const int tid = threadIdx.x;
    const int row = blockIdx.x;

    // Row byte offset: max 4095*100000 = 409.5 MB < 2^31, safe as i32 voffset.
    const unsigned rowByte = (unsigned)row * (unsigned)(V_LEN * 4);

    const uint64_t pBase = (uint64_t)(uintptr_t)p;
    const uint64_t qBase = (uint64_t)(uintptr_t)q;

    // Lane's 16B slot in buffer 0; buffer b is at +b*TILE_B
    // (low 32 bits of the generic pointer == workgroup-relative LDS address).
    const unsigned lp0 = (unsigned)(uintptr_t)&ldsP[0][tid * 4];
    const unsigned lq0 = (unsigned)(uintptr_t)&ldsQ[0][tid * 4];
    const unsigned laneB = (unsigned)tid * 16u;

    const size_t rowBase = (size_t)row * V_LEN;
    const float p0 = p[rowBase];        // needed for the argmax!=0 mask

    float acc  = 0.0f;                  // sum q*(log q - log p) over my lanes' slots
    float vmax = 0.0f;                  // max of p over my lanes' slots (p > 0)

    // Prologue: tiles 0..2 -> buffers 0..2 (6 async ops outstanding)
    #pragma unroll
    for (int t = 0; t < PIPE - 1; ++t) {
        const unsigned voff = rowByte + (unsigned)t * TILE_B + laneB;
        async_ld16(pBase, voff, lp0 + (unsigned)t * TILE_B);
        async_ld16(qBase, voff, lq0 + (unsigned)t * TILE_B);
    }

    for (int t = 0; t < NT; ++t) {
        const unsigned bufOff = (unsigned)(t & (PIPE - 1)) * TILE_B;

        if (t + 3 < NT) {
            const unsigned voff = rowByte + (unsigned)(t + 3) * TILE_B + laneB;
            const unsigned lb   = (unsigned)((t + 3) & (PIPE - 1)) * TILE_B;
            // WAR: ds_loads from the buffer being overwritten (consumed at
            // iteration t-1) must have retired before the async engine writes it.
            asm volatile("s_wait_dscnt 0x0" ::: "memory");
            async_ld16(pBase, voff, lp0 + lb);
            async_ld16(qBase, voff, lq0 + lb);
            // Async loads complete in order per wave: <=6 outstanding means
            // tile t's two loads have landed in LDS (3 tiles ahead in flight).
            asm volatile("s_wait_asynccnt 0x6" ::: "memory");
        } else if (t + 2 < NT) {
            asm volatile("s_wait_asynccnt 0x4" ::: "memory");
        } else if (t + 1 < NT) {
            asm volatile("s_wait_asynccnt 0x2" ::: "memory");
        } else {
            asm volatile("s_wait_asynccnt 0x0" ::: "memory");
        }

        const float4 pv = *reinterpret_cast<const float4*>(
            (const char*)&ldsP[0][0] + bufOff + laneB);
        const float4 qv = *reinterpret_cast<const float4*>(
            (const char*)&ldsQ[0][0] + bufOff + laneB);

        acc += qv.x * (__logf(qv.x) - __logf(pv.x));
        acc += qv.y * (__logf(qv.y) - __logf(pv.y));
        acc += qv.z * (__logf(qv.z) - __logf(pv.z));
        acc += qv.w * (__logf(qv.w) - __logf(pv.w));
        vmax = fmaxf(vmax, fmaxf(fmaxf(pv.x, pv.y), fmaxf(pv.z, pv.w)));
    }

    // Ragged tail: elements [24576, 25000)
    for (int i = REM_OFF + tid; i < V_LEN; i += TPB) {
        const float pe = p[rowBase + i];
        const float qe = q[rowBase + i];
        acc += qe * (__logf(qe) - __logf(pe));
        vmax = fmaxf(vmax, pe);
    }

    // Fixed-order block reduction (deterministic)
    redS[tid] = acc;
    redM[tid] = vmax;
    __syncthreads();
    for (int s = TPB / 2; s > 0; s >>= 1) {
        if (tid < s) {
            redS[tid] += redS[tid + s];
            redM[tid]  = fmaxf(redM[tid], redM[tid + s]);
        }
        __syncthreads();
    }
    if (tid == 0) {
        // argmax over vocab != 0  <=>  p[0] is strictly below the row max
        const float mask = (p0 < redM[0]) ? 1.0f : 0.0f;
        ws[row] = mask * redS[0];
    }
}

__global__ __launch_bounds__(TPB) void final_reduce(const float* __restrict__ ws,
                                                    float* __restrict__ out,
                                                    int rows)
{
    __shared__ float s[TPB];
    const int tid = threadIdx.x;
    float a = 0.0f;
    for (int i = tid; i < rows; i += TPB) a += ws[i];   // fixed order
    s[tid] = a;
    __syncthreads();
    for (int k = TPB / 2; k > 0; k >>= 1) {
        if (tid < k) s[tid] += s[tid + k];
        __syncthreads();
    }
    if (tid == 0) out[0] = s[0] / (float)rows;
}

extern "C" void kernel_launch(void* const* d_in, const int* in_sizes, int n_in,
                              void* d_out, int out_size, void* d_ws, size_t ws_size,
                              hipStream_t stream)
{
    const float* p = (const float*)d_in[0];
    const float* q = (const float*)d_in[1];
    float* out = (float*)d_out;
    float* ws  = (float*)d_ws;                 // needs rows*4 bytes (16 KB)

    const int rows = in_sizes[0] / V_LEN;      // 64*64 = 4096

    kl_row_kernel<<<rows, TPB, 0, stream>>>(p, q, ws);
    final_reduce<<<1, TPB, 0, stream>>>(ws, out, rows);
}